// SketchDecoder_12695923327137
// MI455X (gfx1250) — compile-verified
//
#include <hip/hip_runtime.h>
#include <hip/hip_bf16.h>
#include <math.h>
#include <stdint.h>

#define HDIM 2048
#define MAXS 1000

typedef float v2f __attribute__((ext_vector_type(2)));
typedef float v8f __attribute__((ext_vector_type(8)));

// ---------------------------------------------------------------------------
// Fast device math helpers (CDNA5 has native V_TANH_F32 / V_RCP_F32 / V_EXP_F32)
// ---------------------------------------------------------------------------
__device__ __forceinline__ float rcp_fast(float x) {
#if __has_builtin(__builtin_amdgcn_rcpf)
    return __builtin_amdgcn_rcpf(x);
#else
    return 1.0f / x;
#endif
}

__device__ __forceinline__ float tanh_f(float x) {
#if __has_builtin(__builtin_amdgcn_tanhf)
    return __builtin_amdgcn_tanhf(x);
#elif __has_builtin(__builtin_amdgcn_tanh_f32)
    return __builtin_amdgcn_tanh_f32(x);
#else
    float e = __expf(-2.0f * fabsf(x));
    float t = (1.0f - e) * rcp_fast(1.0f + e);
    return copysignf(t, x);
#endif
}

__device__ __forceinline__ float sigmoid_f(float x) {
#if __has_builtin(__builtin_amdgcn_tanhf) || __has_builtin(__builtin_amdgcn_tanh_f32)
    return fmaf(0.5f, tanh_f(0.5f * x), 0.5f);   // 1/(1+e^-x) = 0.5*tanh(x/2)+0.5
#else
    return rcp_fast(1.0f + __expf(-x));
#endif
}

// ---------------------------------------------------------------------------
// JAX-compatible threefry2x32 (20 rounds) + normal via erfinv
// key(42) -> (k0,k1) = (0,42); bits for shape (2000,) use counters
// x0 = 0..999, x1 = 1000..1999 ; flat = concat(out0, out1), reshape (1000,2).
// ---------------------------------------------------------------------------
struct U2 { uint32_t x, y; };

__device__ __forceinline__ uint32_t rotl32(uint32_t v, int r) {
    return (v << r) | (v >> (32 - r));
}

__device__ U2 threefry2x32(uint32_t k0, uint32_t k1, uint32_t x0, uint32_t x1) {
    const uint32_t ks0 = k0, ks1 = k1, ks2 = 0x1BD11BDAu ^ k0 ^ k1;
    const int rA[4] = {13, 15, 26, 6};
    const int rB[4] = {17, 29, 16, 24};
    x0 += ks0; x1 += ks1;
#pragma unroll
    for (int r = 0; r < 4; ++r) { x0 += x1; x1 = rotl32(x1, rA[r]); x1 ^= x0; }
    x0 += ks1; x1 += ks2 + 1u;
#pragma unroll
    for (int r = 0; r < 4; ++r) { x0 += x1; x1 = rotl32(x1, rB[r]); x1 ^= x0; }
    x0 += ks2; x1 += ks0 + 2u;
#pragma unroll
    for (int r = 0; r < 4; ++r) { x0 += x1; x1 = rotl32(x1, rA[r]); x1 ^= x0; }
    x0 += ks0; x1 += ks1 + 3u;
#pragma unroll
    for (int r = 0; r < 4; ++r) { x0 += x1; x1 = rotl32(x1, rB[r]); x1 ^= x0; }
    x0 += ks1; x1 += ks2 + 4u;
#pragma unroll
    for (int r = 0; r < 4; ++r) { x0 += x1; x1 = rotl32(x1, rA[r]); x1 ^= x0; }
    x0 += ks2; x1 += ks0 + 5u;
    U2 out; out.x = x0; out.y = x1; return out;
}

__device__ float erfinv_f(float x) {
    float w = -__logf((1.0f - x) * (1.0f + x));
    float p;
    if (w < 5.0f) {
        w = w - 2.5f;
        p =  2.81022636e-08f;
        p =  3.43273939e-07f + p * w;
        p = -3.5233877e-06f  + p * w;
        p = -4.39150654e-06f + p * w;
        p =  0.00021858087f  + p * w;
        p = -0.00125372503f  + p * w;
        p = -0.00417768164f  + p * w;
        p =  0.246640727f    + p * w;
        p =  1.50140941f     + p * w;
    } else {
        w = sqrtf(w) - 3.0f;
        p = -0.000200214257f;
        p =  0.000100950558f + p * w;
        p =  0.00134934322f  + p * w;
        p = -0.00367342844f  + p * w;
        p =  0.00573950773f  + p * w;
        p = -0.0076224613f   + p * w;
        p =  0.00943887047f  + p * w;
        p =  1.00167406f     + p * w;
        p =  2.83297682f     + p * w;
    }
    return p * x;
}

__device__ float normal_from_bits(uint32_t b) {
    // uniform in [1,2) -> [0,1) -> [nextafter(-1,0), 1)
    float f = __uint_as_float((b >> 9) | 0x3f800000u) - 1.0f;
    const float lo = -0.99999994f;  // nextafterf(-1,0)
    float u = fmaxf(lo, f * (1.0f - lo) + lo);
    return 1.4142135623730951f * erfinv_f(u);
}

__global__ void eps_kernel(float* __restrict__ eps) {
    int i = blockIdx.x * blockDim.x + threadIdx.x;
    if (i >= MAXS) return;
    uint32_t f0, f1;
    if (i < MAXS / 2) {
        // flat[2i], flat[2i+1] come from out0 (first outputs)
        U2 r0 = threefry2x32(0u, 42u, (uint32_t)(2 * i),     (uint32_t)(MAXS + 2 * i));
        U2 r1 = threefry2x32(0u, 42u, (uint32_t)(2 * i + 1), (uint32_t)(MAXS + 2 * i + 1));
        f0 = r0.x; f1 = r1.x;
    } else {
        // flat[2i] = out1[2i-1000] from counter pair (2i-1000, 2i)
        U2 r0 = threefry2x32(0u, 42u, (uint32_t)(2 * i - MAXS),     (uint32_t)(2 * i));
        U2 r1 = threefry2x32(0u, 42u, (uint32_t)(2 * i + 1 - MAXS), (uint32_t)(2 * i + 1));
        f0 = r0.y; f1 = r1.y;
    }
    eps[2 * i]     = normal_from_bits(f0);
    eps[2 * i + 1] = normal_from_bits(f1);
}

// ---------------------------------------------------------------------------
// Output buffer init: row0 = [0,0,1,0,0], rows 1..999 = [0,0,0,0,1]
// ---------------------------------------------------------------------------
__global__ void init_out_kernel(float* __restrict__ out) {
    int idx = blockIdx.x * blockDim.x + threadIdx.x;
    if (idx >= MAXS * 5) return;
    int r = idx / 5, k = idx % 5;
    float v = (r == 0) ? ((k == 2) ? 1.0f : 0.0f)
                       : ((k == 4) ? 1.0f : 0.0f);
    out[idx] = v;
}

// ---------------------------------------------------------------------------
// Sequential decoder: 1 block x 256 threads (8 wave32s on one WGP).
// Bug-faithful: W_hh never read (hidden state is zero every step).
// gates = W_ih @ last + (b_ih + b_hh); only i/g/o blocks needed.
// W_ih[:,0:4] mat-vec via V_WMMA_F32_16X16X4_F32 with bias passed as the
// C accumulator; col 4 via FMA. Fully straight-line inner loop (EXEC all-1s).
// ---------------------------------------------------------------------------
__global__ __launch_bounds__(256)
void decoder_kernel(const float* __restrict__ Wih,
                    const float* __restrict__ bih,
                    const float* __restrict__ bhh,
                    const float* __restrict__ Wlin,
                    const float* __restrict__ blin,
                    const float* __restrict__ eps,
                    float* __restrict__ out) {
    __shared__ __align__(32) float s_bias_i[HDIM];
    __shared__ __align__(32) float s_bias_g[HDIM];
    __shared__ __align__(32) float s_bias_o[HDIM];
    __shared__ __align__(32) float s_w4_i[HDIM];
    __shared__ __align__(32) float s_w4_g[HDIM];
    __shared__ __align__(32) float s_w4_o[HDIM];
    __shared__ float s_out[8];
    __shared__ float s_last[8];
    __shared__ int   s_ctrl[2];  // [0]=n, [1]=done

    const int tid  = threadIdx.x;
    const int wave = tid >> 5;
    const int lane = tid & 31;
    const int half = lane >> 4;      // 0: K=0,1 / rows m..m+7 ; 1: K=2,3 / rows m+8..m+15
    const int mr   = lane & 15;
    const int k0   = half ? 2 : 0;

    // Precompute combined biases and column-4 weights for i/g/o gate blocks.
    for (int m = tid; m < HDIM; m += 256) {
        s_bias_i[m] = bih[m]            + bhh[m];
        s_bias_g[m] = bih[2 * HDIM + m] + bhh[2 * HDIM + m];
        s_bias_o[m] = bih[3 * HDIM + m] + bhh[3 * HDIM + m];
        s_w4_i[m]   = Wih[(m) * 5 + 4];
        s_w4_g[m]   = Wih[(2 * HDIM + m) * 5 + 4];
        s_w4_o[m]   = Wih[(3 * HDIM + m) * 5 + 4];
    }
    if (tid == 0) {
        s_ctrl[0] = 1; s_ctrl[1] = 0;
        s_last[0] = 0.0f; s_last[1] = 0.0f; s_last[2] = 1.0f;
        s_last[3] = 0.0f; s_last[4] = 0.0f;
    }

    // Preload A tiles (loop-invariant): wave w owns hidden rows [w*256, w*256+256).
    // A layout (32-bit 16x4): lanes 0-15 hold M=lane with VGPR0=K0, VGPR1=K0+1;
    // K0 = 0 (lanes 0-15) or 2 (lanes 16-31).
    v2f a_i[16], a_g[16], a_o[16];
#pragma unroll
    for (int t = 0; t < 16; ++t) {
        int m = wave * 256 + t * 16 + mr;
        const float* pi = Wih + (size_t)(m) * 5 + k0;
        const float* pg = Wih + (size_t)(2 * HDIM + m) * 5 + k0;
        const float* po = Wih + (size_t)(3 * HDIM + m) * 5 + k0;
        a_i[t].x = pi[0]; a_i[t].y = pi[1];
        a_g[t].x = pg[0]; a_g[t].y = pg[1];
        a_o[t].x = po[0]; a_o[t].y = po[1];
    }

    // W_lin partial: column (lane&15) accumulates output j = column if < 7.
    // Clamp the row pointer so every lane loads in-bounds; mask after the loop.
    const bool  jact  = (mr < 7);
    const float* wrow = Wlin + (size_t)(jact ? mr : 0) * HDIM;
    __syncthreads();

    for (int i = 1; i < MAXS; ++i) {
        if (tid < 7) s_out[tid] = 0.0f;
        __syncthreads();

        const float l0 = s_last[0], l1 = s_last[1], l2 = s_last[2],
                    l3 = s_last[3], l4 = s_last[4];
        // B (4x16) with all 16 columns equal to last[0:4):
        // VGPR0 holds K=0 (lanes 0-15) / K=2 (lanes 16-31); VGPR1 holds K=1 / K=3.
        v2f bmat;
        bmat.x = half ? l2 : l0;
        bmat.y = half ? l3 : l1;

        float partial = 0.0f;
#pragma unroll
        for (int t = 0; t < 16; ++t) {
            const int mbase = wave * 256 + t * 16;
            // This lane's 8 output rows (C/D layout: M = v + 8*half, any column).
            const int rbase = mbase + half * 8;

            // Vectorized operand fetch: 2x ds_load_b128 each (32B aligned).
            const v8f cbi = *(const v8f*)&s_bias_i[rbase];
            const v8f cbg = *(const v8f*)&s_bias_g[rbase];
            const v8f cbo = *(const v8f*)&s_bias_o[rbase];
            const v8f wvi = *(const v8f*)&s_w4_i[rbase];
            const v8f wvg = *(const v8f*)&s_w4_g[rbase];
            const v8f wvo = *(const v8f*)&s_w4_o[rbase];
            const v8f wl  = *(const v8f*)(wrow + rbase);

            // D = A x B + bias (bias rides the C accumulator).
            v8f di = __builtin_amdgcn_wmma_f32_16x16x4_f32(
                false, a_i[t], false, bmat, (short)0, cbi, false, false);
            v8f dg = __builtin_amdgcn_wmma_f32_16x16x4_f32(
                false, a_g[t], false, bmat, (short)0, cbg, false, false);
            v8f dq = __builtin_amdgcn_wmma_f32_16x16x4_f32(
                false, a_o[t], false, bmat, (short)0, cbo, false, false);

#pragma unroll
            for (int v = 0; v < 8; ++v) {
                const float ig = fmaf(wvi[v], l4, di[v]);
                const float gg = fmaf(wvg[v], l4, dg[v]);
                const float og = fmaf(wvo[v], l4, dq[v]);
                const float c = sigmoid_f(ig) * tanh_f(gg);
                const float h = sigmoid_f(og) * tanh_f(c);
                partial = fmaf(wl[v], h, partial);
            }
        }
        // Kill lanes whose column is not a real output row, combine row-halves
        // (lane j gets lane j+16), then cross-wave LDS float atomics.
        partial = jact ? partial : 0.0f;
        partial += __shfl_down(partial, 16, 32);
        if (lane < 7) atomicAdd(&s_out[lane], partial);
        __syncthreads();

        if (tid == 0) {
            float o[7];
#pragma unroll
            for (int j = 0; j < 7; ++j) o[j] = tanh_f(s_out[j] + blin[j]);
            const float sig_x = __expf(o[1] * 0.5f);
            const float sig_y = __expf(o[3] * 0.5f);
            const float mx = fmaxf(o[4], fmaxf(o[5], o[6]));
            const float e0 = __expf(o[4] - mx);
            const float e1 = __expf(o[5] - mx);
            const float e2 = __expf(o[6] - mx);
            const float inv = rcp_fast(e0 + e1 + e2);
            const float sx = o[0] + sig_x * eps[2 * i];
            const float sy = o[2] + sig_y * eps[2 * i + 1];
            if (!s_ctrl[1]) {
                const int n = s_ctrl[0];
                float row[5] = {sx, sy, e0 * inv, e1 * inv, e2 * inv};
#pragma unroll
                for (int k = 0; k < 5; ++k) {
                    out[(size_t)n * 5 + k] = row[k];
                    s_last[k] = row[k];
                }
                s_ctrl[0] = n + 1;
                // argmax(new_row[2:]) == 2  <=>  o6 strictly beats o4 and o5
                s_ctrl[1] = (o[6] > o[4] && o[6] > o[5]) ? 1 : 0;
            }
        }
        __syncthreads();
        if (s_ctrl[1]) break;   // uniform: reference iterations past 'done' are no-ops
    }
}

extern "C" void kernel_launch(void* const* d_in, const int* in_sizes, int n_in,
                              void* d_out, int out_size, void* d_ws, size_t ws_size,
                              hipStream_t stream) {
    const float* Wih  = (const float*)d_in[0];   // (4H, 5)
    // d_in[1] = W_hh (4H, H): never read (reference bug keeps hidden state zero)
    const float* bih  = (const float*)d_in[2];   // (4H,)
    const float* bhh  = (const float*)d_in[3];   // (4H,)
    const float* Wlin = (const float*)d_in[4];   // (7, H)
    const float* blin = (const float*)d_in[5];   // (7,)
    float* out = (float*)d_out;                  // (1000, 5)
    float* eps = (float*)d_ws;                   // 2000 floats

    init_out_kernel<<<(MAXS * 5 + 255) / 256, 256, 0, stream>>>(out);
    eps_kernel<<<(MAXS + 255) / 256, 256, 0, stream>>>(eps);
    decoder_kernel<<<1, 256, 0, stream>>>(Wih, bih, bhh, Wlin, blin, eps, out);
}